// EdgeRes_GENLayerWraaper_46016279610082
// MI455X (gfx1250) — compile-verified
//
#include <hip/hip_runtime.h>
#include <hip/hip_bf16.h>

#define N_NODES 50000
#define D 96
#define H 192
#define EPS_MSG 1e-7f
#define LN_EPS 1e-5f
#define MT 32           // nodes per block in the MLP kernel

typedef __attribute__((ext_vector_type(2))) float v2f;
typedef __attribute__((ext_vector_type(8))) float v8f;

// ---- ordered-uint encoding for float atomicMax ----
__device__ __forceinline__ unsigned int enc_f32(float f) {
    unsigned int u = __float_as_uint(f);
    return (u & 0x80000000u) ? ~u : (u | 0x80000000u);
}
__device__ __forceinline__ float dec_f32(unsigned int u) {
    return __uint_as_float((u & 0x80000000u) ? (u & 0x7FFFFFFFu) : ~u);
}
#define ENC_NEG_INF 0x007FFFFFu   // enc(-inf)

// ---------------- kernel 0: init workspace ----------------
__global__ void fill_ws_kernel(unsigned int* __restrict__ segmax,
                               float* __restrict__ denom,
                               float* __restrict__ num, int nd) {
    int gid = blockIdx.x * blockDim.x + threadIdx.x;
    if (gid < nd) {
        segmax[gid] = ENC_NEG_INF;
        denom[gid] = 0.0f;
        num[gid] = 0.0f;
    }
}

// ---------------- kernel 1: segment max over dst ----------------
// one thread = 4 channels of one edge
__global__ __launch_bounds__(256) void edge_max_kernel(
    const float* __restrict__ ea, const int* __restrict__ ei,
    const float* __restrict__ t, unsigned int* __restrict__ segmax, int E) {
    int gid = blockIdx.x * blockDim.x + threadIdx.x;
    if (gid >= E * (D / 4)) return;
    int e = gid / (D / 4);
    int c = (gid % (D / 4)) * 4;
    float tv = t[0];
    float4 v = *(const float4*)(ea + (size_t)e * D + c);
    int d = ei[E + e];   // dst
    unsigned int* p = segmax + (size_t)d * D + c;
    float m0 = fmaxf(v.x, 0.f) + EPS_MSG;
    float m1 = fmaxf(v.y, 0.f) + EPS_MSG;
    float m2 = fmaxf(v.z, 0.f) + EPS_MSG;
    float m3 = fmaxf(v.w, 0.f) + EPS_MSG;
    atomicMax(p + 0, enc_f32(m0 * tv));
    atomicMax(p + 1, enc_f32(m1 * tv));
    atomicMax(p + 2, enc_f32(m2 * tv));
    atomicMax(p + 3, enc_f32(m3 * tv));
}

// ---------------- kernel 2: e=exp(s-max); accumulate denom & msg*e ----------------
__global__ __launch_bounds__(256) void edge_exp_kernel(
    const float* __restrict__ ea, const int* __restrict__ ei,
    const float* __restrict__ t, const unsigned int* __restrict__ segmax,
    float* __restrict__ denom, float* __restrict__ num, int E) {
    int gid = blockIdx.x * blockDim.x + threadIdx.x;
    if (gid >= E * (D / 4)) return;
    int e = gid / (D / 4);
    int c = (gid % (D / 4)) * 4;
    float tv = t[0];
    float4 v = *(const float4*)(ea + (size_t)e * D + c);
    int d = ei[E + e];   // dst
    size_t off = (size_t)d * D + c;
    const unsigned int* mp = segmax + off;
    float msg[4];
    msg[0] = fmaxf(v.x, 0.f) + EPS_MSG;
    msg[1] = fmaxf(v.y, 0.f) + EPS_MSG;
    msg[2] = fmaxf(v.z, 0.f) + EPS_MSG;
    msg[3] = fmaxf(v.w, 0.f) + EPS_MSG;
#pragma unroll
    for (int i = 0; i < 4; ++i) {
        float mx = dec_f32(mp[i]);
        float ex = __expf(msg[i] * tv - mx);
        atomicAdd(denom + off + i, ex);
        atomicAdd(num + off + i, msg[i] * ex);
    }
}

// ---------------- kernel 3: fused node MLP (WMMA f32 16x16x4) ----------------
// agg = num/denom ; hpre = agg@W1+b1 ; LN ; ReLU ; x = ln@W2+b2
__global__ __launch_bounds__(256) void node_mlp_kernel(
    const float* __restrict__ num, const float* __restrict__ denom,
    const float* __restrict__ W1, const float* __restrict__ b1,
    const float* __restrict__ gamma, const float* __restrict__ beta,
    const float* __restrict__ W2, const float* __restrict__ b2,
    float* __restrict__ x, int n_nodes) {
    __shared__ float lds_a[MT][D];        // agg tile (32x96)
    __shared__ float lds_h[MT][H + 4];    // hpre / ln tile (32x192, padded)

    int tid = threadIdx.x;
    int wave = tid >> 5;
    int lane = tid & 31;
    int lane_lo = lane & 15;
    int kh = lane >> 4;                   // lane half: K offset 2*kh
    int base = blockIdx.x * MT;

    // stage agg = num/denom into LDS (coalesced, wave-uniform trip count)
    for (int i = tid; i < MT * D; i += 256) {
        int r = i / D, c = i % D;
        int row = base + r;
        float v = 0.f;
        if (row < n_nodes) {
            float dn = denom[(size_t)row * D + c];
            float nm = num[(size_t)row * D + c];
            v = (dn > 0.f) ? (nm / dn) : 0.f;
        }
        lds_a[r][c] = v;
    }
    __syncthreads();

    // GEMM1: hpre[32][192] = agg @ W1 + b1 ; tiles 2(M) x 12(N), round-robin over 8 waves
    for (int t3 = wave; t3 < 24; t3 += 8) {
        int mt = t3 / 12, nt = t3 % 12;
        int n = nt * 16 + lane_lo;
        v8f acc = {};
        for (int ks = 0; ks < D / 4; ++ks) {
            int k0 = ks * 4 + 2 * kh;
            v2f a, b;
            a.x = lds_a[mt * 16 + lane_lo][k0];
            a.y = lds_a[mt * 16 + lane_lo][k0 + 1];
            b.x = W1[(size_t)k0 * H + n];
            b.y = W1[(size_t)(k0 + 1) * H + n];
            acc = __builtin_amdgcn_wmma_f32_16x16x4_f32(
                false, a, false, b, (short)0, acc, false, false);
        }
        float bias = b1[n];
#pragma unroll
        for (int i = 0; i < 8; ++i) {
            lds_h[mt * 16 + i + 8 * kh][n] = acc[i] + bias;
        }
    }
    __syncthreads();

    // LayerNorm + ReLU: 8 lanes per row, 24 elems each, shfl_xor tree reduce
    {
        int r = tid >> 3;   // 0..31
        int j = tid & 7;    // 0..7
        float s = 0.f, ss = 0.f;
#pragma unroll
        for (int i = 0; i < 24; ++i) {
            float v = lds_h[r][j * 24 + i];
            s += v;
            ss += v * v;
        }
        for (int off = 1; off < 8; off <<= 1) {
            s += __shfl_xor(s, off, 8);
            ss += __shfl_xor(ss, off, 8);
        }
        float mu = s * (1.0f / H);
        float var = ss * (1.0f / H) - mu * mu;
        float inv = rsqrtf(var + LN_EPS);
#pragma unroll
        for (int i = 0; i < 24; ++i) {
            int c = j * 24 + i;
            float v = (lds_h[r][c] - mu) * inv * gamma[c] + beta[c];
            lds_h[r][c] = fmaxf(v, 0.f);
        }
    }
    __syncthreads();

    // GEMM2: x[32][96] = ln @ W2 + b2 ; tiles 2(M) x 6(N)
    for (int t3 = wave; t3 < 12; t3 += 8) {
        int mt = t3 / 6, nt = t3 % 6;
        int n = nt * 16 + lane_lo;
        v8f acc = {};
        for (int ks = 0; ks < H / 4; ++ks) {
            int k0 = ks * 4 + 2 * kh;
            v2f a, b;
            a.x = lds_h[mt * 16 + lane_lo][k0];
            a.y = lds_h[mt * 16 + lane_lo][k0 + 1];
            b.x = W2[(size_t)k0 * D + n];
            b.y = W2[(size_t)(k0 + 1) * D + n];
            acc = __builtin_amdgcn_wmma_f32_16x16x4_f32(
                false, a, false, b, (short)0, acc, false, false);
        }
        float bias = b2[n];
#pragma unroll
        for (int i = 0; i < 8; ++i) {
            int row = base + mt * 16 + i + 8 * kh;
            if (row < n_nodes) x[(size_t)row * D + n] = acc[i] + bias;
        }
    }
}

// ---------------- kernel 4: out = edge_attr + x[src] + x[dst] ----------------
__global__ __launch_bounds__(256) void edge_out_kernel(
    const float* __restrict__ ea, const int* __restrict__ ei,
    const float* __restrict__ x, float* __restrict__ out, int E) {
    int gid = blockIdx.x * blockDim.x + threadIdx.x;
    if (gid >= E * (D / 4)) return;
    int e = gid / (D / 4);
    int c = (gid % (D / 4)) * 4;
    int s = ei[e];       // src
    int d = ei[E + e];   // dst
    float4 v = *(const float4*)(ea + (size_t)e * D + c);
    float4 xs = *(const float4*)(x + (size_t)s * D + c);
    float4 xd = *(const float4*)(x + (size_t)d * D + c);
    float4 o;
    o.x = v.x + xs.x + xd.x;
    o.y = v.y + xs.y + xd.y;
    o.z = v.z + xs.z + xd.z;
    o.w = v.w + xs.w + xd.w;
    *(float4*)(out + (size_t)e * D + c) = o;
}

extern "C" void kernel_launch(void* const* d_in, const int* in_sizes, int n_in,
                              void* d_out, int out_size, void* d_ws, size_t ws_size,
                              hipStream_t stream) {
    const float* edge_attr = (const float*)d_in[0];
    const int* edge_index  = (const int*)d_in[1];
    const float* t         = (const float*)d_in[2];
    const float* W1        = (const float*)d_in[3];
    const float* b1        = (const float*)d_in[4];
    const float* gamma     = (const float*)d_in[5];
    const float* beta      = (const float*)d_in[6];
    const float* W2        = (const float*)d_in[7];
    const float* b2        = (const float*)d_in[8];
    float* out = (float*)d_out;

    int E = in_sizes[1] / 2;
    const int ND = N_NODES * D;

    // workspace layout: [segmax_enc | denom | num]; x aliases segmax once dead
    unsigned int* segmax = (unsigned int*)d_ws;
    float* denom = (float*)d_ws + ND;
    float* num   = (float*)d_ws + 2 * (size_t)ND;
    float* x     = (float*)d_ws;   // reuse

    int fill_blocks = (ND + 255) / 256;
    fill_ws_kernel<<<fill_blocks, 256, 0, stream>>>(segmax, denom, num, ND);

    int et = E * (D / 4);
    int edge_blocks = (et + 255) / 256;
    edge_max_kernel<<<edge_blocks, 256, 0, stream>>>(edge_attr, edge_index, t, segmax, E);
    edge_exp_kernel<<<edge_blocks, 256, 0, stream>>>(edge_attr, edge_index, t, segmax,
                                                     denom, num, E);
    int node_blocks = (N_NODES + MT - 1) / MT;
    node_mlp_kernel<<<node_blocks, 256, 0, stream>>>(num, denom, W1, b1, gamma, beta,
                                                     W2, b2, x, N_NODES);
    edge_out_kernel<<<edge_blocks, 256, 0, stream>>>(edge_attr, edge_index, x, out, E);
}